// PMPA1_ave_52527450030478
// MI455X (gfx1250) — compile-verified
//
#include <hip/hip_runtime.h>
#include <hip/hip_bf16.h>

typedef float v2f __attribute__((ext_vector_type(2)));
typedef float v8f __attribute__((ext_vector_type(8)));

#define WMMA_F32(a, b, c) \
  __builtin_amdgcn_wmma_f32_16x16x4_f32(false, (a), false, (b), (short)0, (c), false, false)

constexpr int Bn = 8, Cn = 32, Hn = 256, Wn = 256;
constexpr int T  = 256;        // tiles per batch (16x16 grid of 16x16 tiles)
constexpr int P  = 256;        // pixels per tile
constexpr int KD = Cn * P;     // 8192 contraction dim

// ---------------------------------------------------------------------------
// Kernel 1: depthwise 3x3 conv (Q and K), scattered into tile-major layout
//   Qt[b][t][c*256 + p],  t = (h/16)*16 + (w/16),  p = (h%16)*16 + (w%16)
// ---------------------------------------------------------------------------
__global__ __launch_bounds__(256) void k_conv_qk(
    const float* __restrict__ x,
    const float* __restrict__ wq, const float* __restrict__ bq,
    const float* __restrict__ wk, const float* __restrict__ bk,
    float* __restrict__ Qt, float* __restrict__ Kt)
{
  int idx = blockIdx.x * 256 + threadIdx.x;
  int w = idx & 255;
  int h = (idx >> 8) & 255;
  int c = (idx >> 16) & 31;
  int b = idx >> 21;
  const float* xc = x + (size_t)(b * Cn + c) * Hn * Wn;
  float q = bq[c], k = bk[c];
#pragma unroll
  for (int kh = 0; kh < 3; ++kh) {
    int hh = h + kh - 1;
    if ((unsigned)hh >= (unsigned)Hn) continue;
#pragma unroll
    for (int kw = 0; kw < 3; ++kw) {
      int ww = w + kw - 1;
      if ((unsigned)ww >= (unsigned)Wn) continue;
      float v = xc[hh * Wn + ww];
      q += v * wq[c * 9 + kh * 3 + kw];
      k += v * wk[c * 9 + kh * 3 + kw];
    }
  }
  int t = ((h >> 4) << 4) + (w >> 4);
  int p = ((h & 15) << 4) + (w & 15);
  size_t off = (size_t)(b * T + t) * KD + c * P + p;
  Qt[off] = q;
  Kt[off] = k;
}

// ---------------------------------------------------------------------------
// Kernel 2: G[b] = Qmat(256x8192) @ Kmat(256x8192)^T via V_WMMA_F32_16X16X4_F32
// One WG per (batch, 16-row i-block). 8 waves, each owns 2 j-blocks.
// ---------------------------------------------------------------------------
constexpr int CK   = 32;       // k-chunk staged in LDS
constexpr int KSTR = CK + 2;   // padded LDS stride (bank spread)

__global__ __launch_bounds__(256) void k_qk_gemm(
    const float* __restrict__ Qt, const float* __restrict__ Kt,
    float* __restrict__ S)
{
  __shared__ float Qs[16 * KSTR];
  __shared__ float Ks[256 * KSTR];

  int b = blockIdx.y, ib = blockIdx.x;
  int tid = threadIdx.x, lane = tid & 31, wv = tid >> 5;
  int lrow = lane & 15, lhi = lane >> 4;

  const float* Qb = Qt + (size_t)(b * T + ib * 16) * KD;
  const float* Kb = Kt + (size_t)b * T * KD;

  int jb0 = wv * 2, jb1 = wv * 2 + 1;
  v8f acc0 = {}; v8f acc1 = {};

  for (int k0 = 0; k0 < KD; k0 += CK) {
    { // stage Q strip: 16 rows x 32
      int r = tid >> 4, cc = (tid & 15) * 2;
      v2f v = *(const v2f*)(Qb + (size_t)r * KD + k0 + cc);
      Qs[r * KSTR + cc] = v.x;
      Qs[r * KSTR + cc + 1] = v.y;
    }
#pragma unroll
    for (int it = 0; it < 8; ++it) { // stage K: 256 rows x 32
      int j = (tid >> 3) + it * 32;
      int cc = (tid & 7) * 4;
      float4 v = *(const float4*)(Kb + (size_t)j * KD + k0 + cc);
      float* d = &Ks[j * KSTR + cc];
      d[0] = v.x; d[1] = v.y; d[2] = v.z; d[3] = v.w;
    }
    __syncthreads();
#pragma unroll
    for (int kk = 0; kk < CK; kk += 4) {
      // A (16x4): lane<16 -> row=lane, K=kk..kk+1; lane>=16 -> K=kk+2..kk+3
      v2f a  = *(const v2f*)&Qs[lrow * KSTR + kk + 2 * lhi];
      // B (4x16) = K^T: same per-lane float2 pattern on K rows
      v2f b0 = *(const v2f*)&Ks[(jb0 * 16 + lrow) * KSTR + kk + 2 * lhi];
      acc0 = WMMA_F32(a, b0, acc0);
      v2f b1 = *(const v2f*)&Ks[(jb1 * 16 + lrow) * KSTR + kk + 2 * lhi];
      acc1 = WMMA_F32(a, b1, acc1);
    }
    __syncthreads();
  }
  // D layout: VGPR r, lane L -> G[i0 + r + 8*(L>>4)][j0 + (L&15)]
  float* Sb = S + (size_t)(b * T + ib * 16) * T;
#pragma unroll
  for (int r = 0; r < 8; ++r) {
    Sb[(r + 8 * lhi) * T + jb0 * 16 + lrow] = acc0[r];
    Sb[(r + 8 * lhi) * T + jb1 * 16 + lrow] = acc1[r];
  }
}

// ---------------------------------------------------------------------------
// Kernel 3: row softmax over j (256 elems). One wave32 per row.
// ---------------------------------------------------------------------------
__global__ __launch_bounds__(256) void k_softmax(float* __restrict__ S)
{
  int row = blockIdx.x * 8 + (threadIdx.x >> 5);
  int lane = threadIdx.x & 31;
  float* Sr = S + (size_t)row * T;
  float v[8];
  float m = -1e30f;
#pragma unroll
  for (int q = 0; q < 8; ++q) { v[q] = Sr[lane + q * 32]; m = fmaxf(m, v[q]); }
#pragma unroll
  for (int off = 16; off > 0; off >>= 1) m = fmaxf(m, __shfl_xor(m, off, 32));
  float s = 0.f;
#pragma unroll
  for (int q = 0; q < 8; ++q) { v[q] = __expf(v[q] - m); s += v[q]; }
#pragma unroll
  for (int off = 16; off > 0; off >>= 1) s += __shfl_xor(s, off, 32);
  float inv = 1.0f / s;
#pragma unroll
  for (int q = 0; q < 8; ++q) Sr[lane + q * 32] = v[q] * inv;
}

// ---------------------------------------------------------------------------
// Kernel 4: Emat[b] = S(256x256) @ Vmat(256x8192), V read directly from x.
// WG per (batch, i-block, 128-col chunk). Each wave: one 16x16 output tile.
// ---------------------------------------------------------------------------
constexpr int SSTR = T + 4;    // 260: 16B-aligned, conflict-free stride

__global__ __launch_bounds__(256) void k_sv_gemm(
    const float* __restrict__ S, const float* __restrict__ x,
    float* __restrict__ Et)
{
  __shared__ float Ss[16 * SSTR];
  int b = blockIdx.z, ib = blockIdx.y, nc = blockIdx.x;
  int tid = threadIdx.x, lane = tid & 31, wv = tid >> 5;
  int lrow = lane & 15, lhi = lane >> 4;

  const float* Sb = S + (size_t)(b * T + ib * 16) * T;
  { // stage the 16 softmax rows (16 x 256)
    int r = tid >> 4, cb = (tid & 15) * 16;
#pragma unroll
    for (int q = 0; q < 4; ++q) {
      float4 v = *(const float4*)(Sb + r * T + cb + q * 4);
      float* d = &Ss[r * SSTR + cb + q * 4];
      d[0] = v.x; d[1] = v.y; d[2] = v.z; d[3] = v.w;
    }
  }
  __syncthreads();

  int n0 = nc * 128 + wv * 16;   // 16 columns, all inside one channel c
  int c  = n0 >> 8;
  int dh = (n0 & 255) >> 4;      // intra-tile pixel row (dw = 0..15 = lane)
  const float* xc = x + (size_t)(b * Cn + c) * Hn * Wn;

  v8f acc = {};
  for (int j0 = 0; j0 < T; j0 += 4) {
    // A (16x4) from S: lane<16 -> (row=lane, j0..j0+1); lane>=16 -> j0+2..j0+3
    v2f a = *(const v2f*)&Ss[lrow * SSTR + j0 + 2 * lhi];
    // B (4x16) = Vmat rows j0..j0+3: lane reads tile jr and jr+1, col lrow
    int jr = j0 + 2 * lhi;
    v2f bv;
    bv.x = xc[(size_t)((((jr    ) >> 4) << 4) + dh) * Wn + (((jr    ) & 15) << 4) + lrow];
    bv.y = xc[(size_t)((((jr + 1) >> 4) << 4) + dh) * Wn + (((jr + 1) & 15) << 4) + lrow];
    acc = WMMA_F32(a, bv, acc);
  }

  float* Eb = Et + (size_t)(b * T + ib * 16) * KD + n0;
#pragma unroll
  for (int r = 0; r < 8; ++r)
    Eb[(size_t)(r + 8 * lhi) * KD + lrow] = acc[r];
}

// ---------------------------------------------------------------------------
// Kernel 5: out[b,o,h,w] = sum_c w1[o,c] * (E[b,tile,c*256+p] + x[b,c,h,w])
// ---------------------------------------------------------------------------
__global__ __launch_bounds__(256) void k_epilogue(
    const float* __restrict__ Et, const float* __restrict__ x,
    const float* __restrict__ w1, float* __restrict__ out)
{
  __shared__ float w1s[Cn * Cn];
  int tid = threadIdx.x;
#pragma unroll
  for (int q = 0; q < 4; ++q) w1s[tid + q * 256] = w1[tid + q * 256];
  __syncthreads();

  int idx = blockIdx.x * 256 + tid;
  int w = idx & 255;
  int h = (idx >> 8) & 255;
  int b = idx >> 16;
  int t = ((h >> 4) << 4) + (w >> 4);
  int p = ((h & 15) << 4) + (w & 15);
  const float* Eb = Et + (size_t)(b * T + t) * KD + p;

  float e[Cn];
#pragma unroll
  for (int c = 0; c < Cn; ++c)
    e[c] = Eb[c * P] + x[((size_t)(b * Cn + c) * Hn + h) * Wn + w];

#pragma unroll 4
  for (int o = 0; o < Cn; ++o) {
    float acc = 0.f;
#pragma unroll
    for (int c = 0; c < Cn; ++c) acc = fmaf(w1s[o * Cn + c], e[c], acc);
    out[((size_t)(b * Cn + o) * Hn + h) * Wn + w] = acc;
  }
}

// ---------------------------------------------------------------------------
extern "C" void kernel_launch(void* const* d_in, const int* in_sizes, int n_in,
                              void* d_out, int out_size, void* d_ws, size_t ws_size,
                              hipStream_t stream)
{
  const float* x  = (const float*)d_in[0];
  const float* wq = (const float*)d_in[1];
  const float* bq = (const float*)d_in[2];
  const float* wk = (const float*)d_in[3];
  const float* bk = (const float*)d_in[4];
  const float* w1 = (const float*)d_in[5];
  float* out = (float*)d_out;

  float* Qt = (float*)d_ws;                 // B*T*KD floats (64 MB)
  float* Kt = Qt + (size_t)Bn * T * KD;     // B*T*KD floats (64 MB)
  float* S  = Kt + (size_t)Bn * T * KD;     // B*T*T  floats (2 MB)
  float* Et = Qt;                           // reuse Qt after QK^T is done

  k_conv_qk<<<(Bn * Cn * Hn * Wn) / 256, 256, 0, stream>>>(x, wq, bq, wk, bk, Qt, Kt);
  k_qk_gemm<<<dim3(16, Bn), 256, 0, stream>>>(Qt, Kt, S);
  k_softmax<<<(Bn * T) / 8, 256, 0, stream>>>(S);
  k_sv_gemm<<<dim3(KD / 128, 16, Bn), 256, 0, stream>>>(S, x, Et);
  k_epilogue<<<(Bn * Hn * Wn) / 256, 256, 0, stream>>>(Et, x, w1, out);
}